// FPModule_52974126629570
// MI455X (gfx1250) — compile-verified
//
#include <hip/hip_runtime.h>
#include <hip/hip_bf16.h>

typedef __attribute__((ext_vector_type(2))) float v2f;
typedef __attribute__((ext_vector_type(8))) float v8f;

#define NB      16
#define N_SRC   1024
#define N_TGT   4096
#define C_IN    256
#define C_SKIP  64
#define C_HID   256
#define M_TOT   (NB * N_TGT)   /* 65536 */
#define A_LD    324            /* 320 padded: 324 % 64 == 4 -> conflict-free */
#define H_LD    260            /* 256 padded: 260 % 64 == 4 -> conflict-free */

static __device__ __forceinline__ v8f vzero8() { v8f z = {}; return z; }

// ---------------------------------------------------------------------------
// Kernel 1: 3-NN search only. Emits absolute source-row indices + normalized
// inverse-distance weights (24 B/target = 1.5 MB total, no feature traffic).
// One block = 256 consecutive targets (same batch); source positions in LDS.
// ---------------------------------------------------------------------------
__global__ void __launch_bounds__(256) knn_kernel(
    const float* __restrict__ pos, const float* __restrict__ pos_t,
    int* __restrict__ ws_idx, float* __restrict__ ws_w)
{
  __shared__ float sp[N_SRC * 3];     // 12 KB source positions

  const int tid = threadIdx.x;
  const int blk = blockIdx.x;
  const int b   = blk >> 4;           // 16 blocks of 256 targets per batch
  const int t0  = blk << 8;

  const float* ps = pos + (size_t)b * N_SRC * 3;
  for (int i = tid; i < N_SRC * 3; i += 256) sp[i] = ps[i];
  __syncthreads();

  const int t = t0 + tid;
  const float tx = pos_t[t * 3 + 0];
  const float ty = pos_t[t * 3 + 1];
  const float tz = pos_t[t * 3 + 2];

  float d0 = 3.4e38f, d1 = 3.4e38f, d2 = 3.4e38f;
  int   i0 = 0, i1 = 0, i2 = 0;
  for (int s = 0; s < N_SRC; ++s) {   // broadcast LDS reads, conflict-free
    const float dx = tx - sp[3 * s + 0];
    const float dy = ty - sp[3 * s + 1];
    const float dz = tz - sp[3 * s + 2];
    const float d  = dx * dx + dy * dy + dz * dz;
    if (d < d2) {
      if (d < d1) {
        d2 = d1; i2 = i1;
        if (d < d0) { d1 = d0; i1 = i0; d0 = d; i0 = s; }
        else        { d1 = d;  i1 = s; }
      } else { d2 = d; i2 = s; }
    }
  }
  const float w0 = 1.0f / fmaxf(d0, 1e-16f);
  const float w1 = 1.0f / fmaxf(d1, 1e-16f);
  const float w2 = 1.0f / fmaxf(d2, 1e-16f);
  const float inv = 1.0f / (w0 + w1 + w2);
  const int base = b * N_SRC;
  ws_idx[t * 3 + 0] = base + i0;  ws_w[t * 3 + 0] = w0 * inv;
  ws_idx[t * 3 + 1] = base + i1;  ws_w[t * 3 + 1] = w1 * inv;
  ws_idx[t * 3 + 2] = base + i2;  ws_w[t * 3 + 2] = w2 * inv;
}

// ---------------------------------------------------------------------------
// Kernel 2: fused  interp-gather + out = relu([xi | x_skip] @ W1 + b1) @ W2 + b2
// Block = 32 output rows, 8 waves (2 along M x 4 along N), fp32 WMMA 16x16x4.
// A tile is produced in LDS directly from gathered x rows (L2-resident).
// ---------------------------------------------------------------------------
__global__ void __launch_bounds__(256) mlp_kernel(
    const float* __restrict__ x,
    const int* __restrict__ ws_idx, const float* __restrict__ ws_w,
    const float* __restrict__ xskip,
    const float* __restrict__ W1, const float* __restrict__ b1,
    const float* __restrict__ W2, const float* __restrict__ b2,
    float* __restrict__ out)
{
  __shared__ __align__(16) float smem[32 * A_LD];  // 41.5 KB, reused for H
  __shared__ int   tIdx[32 * 3];
  __shared__ float tW[32 * 3];

  const int tid   = threadIdx.x;
  const int lane  = tid & 31;
  const int wave  = tid >> 5;
  const int wm    = wave & 1;        // M sub-tile (0..1)
  const int wn    = wave >> 1;       // N sub-tile (0..3)
  const int row0  = blockIdx.x * 32;
  const int l15   = lane & 15;
  const int khalf = (lane >> 4) << 1;  // lanes 0-15 -> K 0/1, 16-31 -> K 2/3
  const int mh    = (lane >> 4) << 3;  // C/D: upper lanes hold M+8

  // ---- per-row KNN metadata for this tile (broadcast table)
  if (tid < 96) {
    tIdx[tid] = ws_idx[row0 * 3 + tid];
    tW[tid]   = ws_w[row0 * 3 + tid];
  }
  __syncthreads();

  // ---- stage A tile [32 x 320] in LDS:
  //      cols 0..255 = 3-row gather-blend of x (coalesced 1KB row reads),
  //      cols 256..319 = x_skip.
  {
    const int c = tid;  // thread == channel
#pragma unroll 4
    for (int r = 0; r < 32; ++r) {
      const int   j0 = tIdx[r * 3 + 0], j1 = tIdx[r * 3 + 1], j2 = tIdx[r * 3 + 2];
      const float a0 = tW[r * 3 + 0],   a1 = tW[r * 3 + 1],   a2 = tW[r * 3 + 2];
      const float v = a0 * x[(size_t)j0 * C_IN + c]
                    + a1 * x[(size_t)j1 * C_IN + c]
                    + a2 * x[(size_t)j2 * C_IN + c];
      smem[r * A_LD + c] = v;
    }
  }
  for (int i = tid; i < 32 * 16; i += 256) {
    const int r = i >> 4, c4 = i & 15;
    const float4 v = ((const float4*)(xskip + (size_t)(row0 + r) * C_SKIP))[c4];
    *(float4*)&smem[r * A_LD + 256 + c4 * 4] = v;
  }
  __syncthreads();

  const int m_base = wm * 16;
  const int n_base = wn * 64;

  v8f acc[4];
#pragma unroll
  for (int j = 0; j < 4; ++j) acc[j] = vzero8();

  // ---- GEMM1: K = 320
  const float* arow = &smem[(m_base + l15) * A_LD + khalf];
  for (int k0 = 0; k0 < C_IN + C_SKIP; k0 += 4) {
    const v2f a = *(const v2f*)(arow + k0);
#pragma unroll
    for (int j = 0; j < 4; ++j) {
      const int n = n_base + j * 16 + l15;
      const float* wp = W1 + (size_t)(k0 + khalf) * C_HID + n;
      v2f bf; bf.x = wp[0]; bf.y = wp[C_HID];
      acc[j] = __builtin_amdgcn_wmma_f32_16x16x4_f32(
          false, a, false, bf, (short)0, acc[j], false, false);
    }
  }
  __syncthreads();  // all LDS A reads complete before overwrite

  // ---- bias + ReLU -> H in LDS (stride 260)
#pragma unroll
  for (int j = 0; j < 4; ++j) {
    const int n = n_base + j * 16 + l15;
    const float bb = b1[n];
#pragma unroll
    for (int r = 0; r < 8; ++r) {
      smem[(m_base + r + mh) * H_LD + n] = fmaxf(acc[j][r] + bb, 0.0f);
    }
    acc[j] = vzero8();
  }
  __syncthreads();

  // ---- GEMM2: K = 256
  const float* hrow = &smem[(m_base + l15) * H_LD + khalf];
  for (int k0 = 0; k0 < C_HID; k0 += 4) {
    const v2f a = *(const v2f*)(hrow + k0);
#pragma unroll
    for (int j = 0; j < 4; ++j) {
      const int n = n_base + j * 16 + l15;
      const float* wp = W2 + (size_t)(k0 + khalf) * C_HID + n;
      v2f bf; bf.x = wp[0]; bf.y = wp[C_HID];
      acc[j] = __builtin_amdgcn_wmma_f32_16x16x4_f32(
          false, a, false, bf, (short)0, acc[j], false, false);
    }
  }

  // ---- bias + store
#pragma unroll
  for (int j = 0; j < 4; ++j) {
    const int n = n_base + j * 16 + l15;
    const float bb = b2[n];
#pragma unroll
    for (int r = 0; r < 8; ++r) {
      const int m = m_base + r + mh;
      out[(size_t)(row0 + m) * C_HID + n] = acc[j][r] + bb;
    }
  }
}

// ---------------------------------------------------------------------------
// Kernel 3: tuple tail — pos_skip passthrough + batch_skip (int -> float)
// ---------------------------------------------------------------------------
__global__ void __launch_bounds__(256) tail_kernel(
    const float* __restrict__ pos_t, const int* __restrict__ batch_t,
    float* __restrict__ out)
{
  const int i = blockIdx.x * 256 + threadIdx.x;
  if (i < M_TOT * 3)
    out[(size_t)M_TOT * C_HID + i] = pos_t[i];
  if (i < M_TOT)
    out[(size_t)M_TOT * C_HID + (size_t)M_TOT * 3 + i] = (float)batch_t[i];
}

extern "C" void kernel_launch(void* const* d_in, const int* in_sizes, int n_in,
                              void* d_out, int out_size, void* d_ws, size_t ws_size,
                              hipStream_t stream) {
  const float* x         = (const float*)d_in[0];
  const float* pos       = (const float*)d_in[1];
  // d_in[2] = batch (unused; layout is implicit)
  const float* x_skip    = (const float*)d_in[3];
  const float* pos_skip  = (const float*)d_in[4];
  const int*   batch_sk  = (const int*)d_in[5];
  const float* W1        = (const float*)d_in[6];
  const float* b1        = (const float*)d_in[7];
  const float* W2        = (const float*)d_in[8];
  const float* b2        = (const float*)d_in[9];
  float* out = (float*)d_out;

  // workspace: 3 ints + 3 floats per target = 1.5 MB total
  int*   ws_idx = (int*)d_ws;
  float* ws_w   = (float*)d_ws + (size_t)3 * M_TOT;

  knn_kernel<<<M_TOT / 256, 256, 0, stream>>>(pos, pos_skip, ws_idx, ws_w);
  mlp_kernel<<<M_TOT / 32, 256, 0, stream>>>(x, ws_idx, ws_w, x_skip,
                                             W1, b1, W2, b2, out);
  tail_kernel<<<(M_TOT * 3 + 255) / 256, 256, 0, stream>>>(pos_skip, batch_sk, out);
}